// MyInterleavedModule_1717986918494
// MI455X (gfx1250) — compile-verified
//
#include <hip/hip_runtime.h>

// ---------------------------------------------------------------------------
// out[n,o] = sum_k x[n,k] * w[o,k]   (N=8192, O=4096, K=4096, fp32)
// Memory-bound: 274 GFLOP vs 336MB unique traffic @ 23.3TB/s => ~14.4us floor.
// Full-precision V_WMMA_F32_16X16X4_F32 (exact fp32 accumulate).
// Block tile 128x128, BK=32, double-buffered LDS.
// Staging: GLOBAL_LOAD_ASYNC_TO_LDS_B128 (ASYNCcnt-tracked, no staging VGPRs).
// ---------------------------------------------------------------------------

typedef __attribute__((ext_vector_type(2))) float v2f;
typedef __attribute__((ext_vector_type(4))) float v4f;
typedef __attribute__((ext_vector_type(8))) float v8f;

// The async-to-LDS builtin takes pointers to a GCC-vector int4 (per clang's
// diagnostic: "__attribute__((__vector_size__(4 * sizeof(int)))) int *").
typedef int v4i_gcc __attribute__((__vector_size__(16)));
typedef __attribute__((address_space(1))) v4i_gcc* gv4p;   // global src
typedef __attribute__((address_space(3))) v4i_gcc* lv4p;   // LDS dst

#define K_DIM   4096
#define OUT_F   4096
#define N_ROWS  8192
#define BM      128
#define BN      128
#define BK      32
#define BKP     36            // padded LDS row stride (floats): 144B -> 16B aligned
                              // and 36*m mod 64 banks distinct for m=0..15

#if __has_builtin(__builtin_amdgcn_global_load_async_to_lds_b128)
#define HAVE_ASYNC_LDS 1
#else
#define HAVE_ASYNC_LDS 0
#endif

__device__ __forceinline__ void wait_async0() {
#if __has_builtin(__builtin_amdgcn_s_wait_asynccnt)
    __builtin_amdgcn_s_wait_asynccnt(0);
#else
    asm volatile("s_wait_asynccnt 0" ::: "memory");
#endif
}

__global__ __launch_bounds__(256)
void gemm_f32_wmma(const float* __restrict__ X,
                   const float* __restrict__ W,
                   float* __restrict__ Out)
{
    __shared__ float As[2][BM * BKP];   // x tile   [m][k]
    __shared__ float Bs[2][BN * BKP];   // w tile   [o][k]  (transpose falls out of layout)

    const int tid  = threadIdx.x;
    const int lane = tid & 31;
    const int wave = tid >> 5;
    const int hi16 = lane >> 4;        // lane-half selects K pair (f32 A/B layout)
    const int l16  = lane & 15;

    const int m0 = blockIdx.y * BM;
    const int n0 = blockIdx.x * BN;

    const int wm = (wave >> 2) * 64;   // 2 waves along M -> 64 rows each
    const int wn = (wave & 3) * 32;    // 4 waves along N -> 32 cols each

    v8f acc[4][2];
#pragma unroll
    for (int mt = 0; mt < 4; ++mt)
#pragma unroll
        for (int nt = 0; nt < 2; ++nt)
            acc[mt][nt] = (v8f)0.0f;

    // global -> LDS staging: 128x32 floats = 1024 float4 per tile, 4 per thread
    auto load_tile = [&](int buf, int kt) {
        const int kbase = kt * BK;
#pragma unroll
        for (int i = 0; i < 4; ++i) {
            const int f4id = tid + i * 256;
            const int row  = f4id >> 3;
            const int c4   = (f4id & 7) << 2;
            const float* ga = X + (size_t)(m0 + row) * K_DIM + kbase + c4;
            const float* gb = W + (size_t)(n0 + row) * K_DIM + kbase + c4;
            float* la = &As[buf][row * BKP + c4];
            float* lb = &Bs[buf][row * BKP + c4];
#if HAVE_ASYNC_LDS
            // GLOBAL_LOAD_ASYNC_TO_LDS_B128: memory -> LDS direct, ASYNCcnt.
            __builtin_amdgcn_global_load_async_to_lds_b128(
                (gv4p)ga, (lv4p)la, /*imm offset*/0, /*cpol*/0);
            __builtin_amdgcn_global_load_async_to_lds_b128(
                (gv4p)gb, (lv4p)lb, /*imm offset*/0, /*cpol*/0);
#else
            const v4f a = *(const v4f*)ga;
            const v4f b = *(const v4f*)gb;
            *(v4f*)la = a;                       // ds_store_b128 (16B aligned)
            *(v4f*)lb = b;
#endif
        }
    };

    load_tile(0, 0);

    const int NK = K_DIM / BK;          // 128 K-tiles
    for (int kt = 0; kt < NK; ++kt) {
        const int cur = kt & 1;
#if HAVE_ASYNC_LDS
        wait_async0();                   // own async writes into `cur` complete
#endif
        __syncthreads();                 // all waves' writes visible, prev reads retired
        if (kt + 1 < NK)
            load_tile(cur ^ 1, kt + 1);  // overlap next staging with compute

        const float* __restrict__ A = &As[cur][0];
        const float* __restrict__ B = &Bs[cur][0];

#pragma unroll
        for (int kk = 0; kk < BK; kk += 4) {
            // A 16x4 f32 fragment: lanes 0-15 -> K=kk,kk+1 ; lanes 16-31 -> K=kk+2,kk+3
            v2f a[4], b[2];
#pragma unroll
            for (int mt = 0; mt < 4; ++mt)
                a[mt] = *(const v2f*)&A[(wm + mt * 16 + l16) * BKP + kk + 2 * hi16];
            // B 4x16 f32 fragment: lane = N (=o), lane-half = K pair; B[k][n] = w[o=n][k]
#pragma unroll
            for (int nt = 0; nt < 2; ++nt)
                b[nt] = *(const v2f*)&B[(wn + nt * 16 + l16) * BKP + kk + 2 * hi16];

#pragma unroll
            for (int mt = 0; mt < 4; ++mt)
#pragma unroll
                for (int nt = 0; nt < 2; ++nt)
                    acc[mt][nt] = __builtin_amdgcn_wmma_f32_16x16x4_f32(
                        /*neg_a=*/false, a[mt],
                        /*neg_b=*/false, b[nt],
                        /*c_mod=*/(short)0, acc[mt][nt],
                        /*reuse_a=*/false, /*reuse_b=*/false);
        }
    }

    // C/D 16x16 f32 layout: VGPR r, lanes 0-15 -> M=r, N=lane ; lanes 16-31 -> M=r+8
#pragma unroll
    for (int mt = 0; mt < 4; ++mt)
#pragma unroll
        for (int nt = 0; nt < 2; ++nt) {
            const int o = n0 + wn + nt * 16 + l16;
#pragma unroll
            for (int r = 0; r < 8; ++r) {
                const int m = m0 + wm + mt * 16 + r + 8 * hi16;
                Out[(size_t)m * OUT_F + o] = acc[mt][nt][r];
            }
        }
}

extern "C" void kernel_launch(void* const* d_in, const int* in_sizes, int n_in,
                              void* d_out, int out_size, void* d_ws, size_t ws_size,
                              hipStream_t stream) {
    const float* x = (const float*)d_in[0];   // (4, 2048, 4096) fp32
    const float* w = (const float*)d_in[1];   // (4096, 4096)    fp32
    float* out = (float*)d_out;               // (4, 2048, 4096) fp32

    dim3 grid(OUT_F / BN, N_ROWS / BM);       // 32 x 64 workgroups
    gemm_f32_wmma<<<grid, 256, 0, stream>>>(x, w, out);
}